// myMSELoss_9835475108015
// MI455X (gfx1250) — compile-verified
//
#include <hip/hip_runtime.h>
#include <hip/hip_bf16.h>

typedef __attribute__((ext_vector_type(2))) float v2f;
typedef __attribute__((ext_vector_type(4))) float f4;
typedef __attribute__((ext_vector_type(8))) float v8f;

// x^255 = x^(2^8 - 1): 7 squarings + 7 products (exact repeated-squaring chain)
__device__ __forceinline__ float pow255(float x) {
    float p = x;          // x^1
    float s = x * x;      // x^2
    p *= s;               // x^3
    s *= s;               // x^4
    p *= s;               // x^7
    s *= s;               // x^8
    p *= s;               // x^15
    s *= s;               // x^16
    p *= s;               // x^31
    s *= s;               // x^32
    p *= s;               // x^63
    s *= s;               // x^64
    p *= s;               // x^127
    s *= s;               // x^128
    p *= s;               // x^255
    return p;
}

// Wave32 sum-reduction through the matrix unit:
// A = ones(16x4), B holds the 32 per-lane partials (other 32 slots = 0).
// D[m][n] = sum_k B[k][n]  (identical for every row m), i.e. one
// v_wmma_f32_16x16x4_f32 collapses the K dimension; 4 xor-shuffles then sum
// the 16 column sums. Every lane ends up holding the wave total.
// MUST be called with EXEC all-1s (top-level, all threads active).
__device__ __forceinline__ float wave_reduce_wmma(float acc) {
    v2f a; a.x = 1.0f; a.y = 1.0f;
    v2f b; b.x = acc;  b.y = 0.0f;
    v8f c = {};
    v8f d = __builtin_amdgcn_wmma_f32_16x16x4_f32(
        /*neg_a=*/false, a, /*neg_b=*/false, b,
        /*c_mod=*/(short)0, c, /*reuse_a=*/false, /*reuse_b=*/false);
    float s = d[0];
    s += __shfl_xor(s, 1);
    s += __shfl_xor(s, 2);
    s += __shfl_xor(s, 4);
    s += __shfl_xor(s, 8);
    return s;
}

// Pass 1: streaming grid-stride over float4s with non-temporal loads
// (read-once data; do not pollute the 192MB L2), per-block partial sums.
__global__ void mse_partial_kernel(const float* __restrict__ img1,
                                   const float* __restrict__ img2,
                                   float* __restrict__ partial,
                                   long long n4) {
    __shared__ float wsum[8];
    const f4* a4 = (const f4*)img1;
    const f4* b4 = (const f4*)img2;

    long long i      = (long long)blockIdx.x * blockDim.x + threadIdx.x;
    long long stride = (long long)gridDim.x * blockDim.x;

    float acc = 0.0f;
    for (; i < n4; i += stride) {
        f4 a = __builtin_nontemporal_load(a4 + i);
        f4 b = __builtin_nontemporal_load(b4 + i);
        float d0 = a.x * 255.0f - pow255(b.x);
        float d1 = a.y * 255.0f - pow255(b.y);
        float d2 = a.z * 255.0f - pow255(b.z);
        float d3 = a.w * 255.0f - pow255(b.w);
        acc = fmaf(d0, d0, acc);
        acc = fmaf(d1, d1, acc);
        acc = fmaf(d2, d2, acc);
        acc = fmaf(d3, d3, acc);
    }

    // Stage 1: matrix-unit wave reduction (all threads active here).
    float w = wave_reduce_wmma(acc);

    const int lane = threadIdx.x & 31;
    const int wid  = threadIdx.x >> 5;
    if (lane == 0) wsum[wid] = w;
    __syncthreads();

    // Stage 2: 8 wave totals -> block total (plain shuffles: EXEC may be
    // masked inside this branch, so no WMMA here).
    if (wid == 0) {
        float v = (lane < 8) ? wsum[lane] : 0.0f;
        v += __shfl_xor(v, 1);
        v += __shfl_xor(v, 2);
        v += __shfl_xor(v, 4);
        if (lane == 0) partial[blockIdx.x] = v;
    }
}

// Pass 2: single block folds the per-block partials deterministically and
// scales by 1/N. No float atomics -> bit-identical result on every replay.
__global__ void mse_final_kernel(const float* __restrict__ partial,
                                 float* __restrict__ out,
                                 int nparts, float inv_n) {
    __shared__ float wsum[8];
    float acc = 0.0f;
    for (int i = threadIdx.x; i < nparts; i += blockDim.x)
        acc += partial[i];

    float w = wave_reduce_wmma(acc);  // all threads active: EXEC all-1s

    const int lane = threadIdx.x & 31;
    const int wid  = threadIdx.x >> 5;
    if (lane == 0) wsum[wid] = w;
    __syncthreads();

    if (wid == 0) {
        float v = (lane < 8) ? wsum[lane] : 0.0f;
        v += __shfl_xor(v, 1);
        v += __shfl_xor(v, 2);
        v += __shfl_xor(v, 4);
        if (lane == 0) out[0] = v * inv_n;
    }
}

extern "C" void kernel_launch(void* const* d_in, const int* in_sizes, int n_in,
                              void* d_out, int out_size, void* d_ws, size_t ws_size,
                              hipStream_t stream) {
    const float* img1 = (const float*)d_in[0];
    const float* img2 = (const float*)d_in[1];
    float* out   = (float*)d_out;
    float* parts = (float*)d_ws;

    const long long n  = (long long)in_sizes[0];  // 64*3*512*512, divisible by 4
    const long long n4 = n >> 2;

    int blocks = 2048;                            // ~24 float4s per thread
    long long maxp = (long long)(ws_size / sizeof(float));
    if (maxp < blocks) blocks = (int)(maxp > 0 ? maxp : 1);

    mse_partial_kernel<<<blocks, 256, 0, stream>>>(img1, img2, parts, n4);
    mse_final_kernel<<<1, 256, 0, stream>>>(parts, out, blocks,
                                            1.0f / (float)n);
}